// GeneralisedShapeletTransform_55903294325333
// MI455X (gfx1250) — compile-verified
//
#include <hip/hip_runtime.h>
#include <math.h>

typedef __attribute__((ext_vector_type(2))) float v2f;
typedef __attribute__((ext_vector_type(8))) float v8f;

namespace {
constexpr int kL   = 4096;  // path length
constexpr int kC   = 8;     // channels
constexpr int kS   = 256;   // NUM_CONTINUOUS_SAMPLES (start positions)
constexpr int kNSh = 16;    // NUM_SHAPELETS
constexpr int kK   = 32;    // NUM_SHAPELET_SAMPLES
}

// Initialize output to +inf bit pattern so positive-float bitwise atomicMin works.
__global__ void shapelet_init_kernel(int* out, int n) {
  int i = blockIdx.x * blockDim.x + threadIdx.x;
  if (i < n) out[i] = 0x7F800000;  // +inf
}

// One wave (32 lanes) per (shapelet, start, batch-half).
// Lane l: batch m = l&15 (+16*half), chunk-element offset off = 2*(l>>4).
// A[m,K] = d values, B[K,n] = h values; diag of D accumulates sum(d*h) per batch.
__global__ __launch_bounds__(32)
void shapelet_wmma_kernel(const float* __restrict__ times,
                          const float* __restrict__ path,
                          const float* __restrict__ lengths,
                          const float* __restrict__ shapelets,
                          int* __restrict__ out_bits) {
  const int lane = threadIdx.x;      // 0..31
  const int half = blockIdx.x;       // 0..1  -> batches [0,16) or [16,32)
  const int s    = blockIdx.y;       // 0..255 start sample
  const int sh   = blockIdx.z;       // 0..15  shapelet

  const int m   = lane & 15;
  const int off = (lane >> 4) << 1;  // 0 (lanes 0-15) or 2 (lanes 16-31)
  const int b   = half * 16 + m;

  float len = lengths[sh];
  len = fminf(fmaxf(len, 0.01f), 512.0f);
  const float t0 = times[0];
  const float tN = times[kL - 1];
  const float start = t0 + ((float)s / (float)(kS - 1)) * (tN - len - t0);
  // dt = len/(K-1); also the per-k query step. One division per wave, hoisted
  // out of the loop so the body has no v_div_* sequence.
  const float dt = len / (float)(kK - 1);

  const float* prow = path + ((size_t)b * kL) * kC;
  const float* wrow = shapelets + ((size_t)sh * kK) * kC;

  v8f acc = {};
  // d at k+1 (per-lane 4 channels: off, off+1, off+4, off+5)
  float pA = 0.0f, pB = 0.0f, pC = 0.0f, pD = 0.0f;

  for (int k = kK - 1; k >= 0; --k) {
    const float q = fmaf((float)k, dt, start);
    int idx = (int)floorf(q);
    idx = idx < 0 ? 0 : (idx > kL - 2 ? kL - 2 : idx);
    const float r = q - (float)idx;  // times is arange -> unit knot spacing

    const float* r0 = prow + (size_t)idx * kC + off;
    const float2 p0lo = *(const float2*)(r0);
    const float2 p0hi = *(const float2*)(r0 + 4);
    const float2 p1lo = *(const float2*)(r0 + kC);
    const float2 p1hi = *(const float2*)(r0 + kC + 4);

    const float* wr = wrow + k * kC + off;
    const float2 wlo = *(const float2*)(wr);
    const float2 whi = *(const float2*)(wr + 4);

    // d = interp(path, q) - shapelet
    const float dA = (p0lo.x + r * (p1lo.x - p0lo.x)) - wlo.x;
    const float dB = (p0lo.y + r * (p1lo.y - p0lo.y)) - wlo.y;
    const float dC = (p0hi.x + r * (p1hi.x - p0hi.x)) - whi.x;
    const float dD = (p0hi.y + r * (p1hi.y - p0hi.y)) - whi.y;

    // h = a_k * d_k + d_{k+1}   (a = 1 at endpoints, 2 inside)
    const float a = (k == 0 || k == kK - 1) ? 1.0f : 2.0f;
    const float hA = fmaf(a, dA, pA);
    const float hB = fmaf(a, dB, pB);
    const float hC = fmaf(a, dC, pC);
    const float hD = fmaf(a, dD, pD);

    // Chunk 2k: channels cbase=0..3 ; this lane holds K=off, off+1.
    {
      v2f A0 = {dA, dB};
      v2f B0 = {hA, hB};
      acc = __builtin_amdgcn_wmma_f32_16x16x4_f32(
          false, A0, false, B0, (short)0, acc, false, false);
    }
    // Chunk 2k+1: channels cbase=4..7.
    {
      v2f A1 = {dC, dD};
      v2f B1 = {hC, hD};
      acc = __builtin_amdgcn_wmma_f32_16x16x4_f32(
          false, A1, false, B1, (short)0, acc, false, false);
    }

    pA = dA; pB = dB; pC = dC; pD = dD;
  }

  // Extract diagonal D[m,m]: VGPR v, lane v -> M=N=v ; VGPR v, lane v+24 -> M=N=v+8.
  const float scale = dt * (1.0f / 3.0f);
  float myval = 0.0f;
  int mybatch = -1;
#pragma unroll
  for (int v = 0; v < 8; ++v) {
    if (lane == v)      { myval = acc[v]; mybatch = half * 16 + v; }
    if (lane == v + 24) { myval = acc[v]; mybatch = half * 16 + v + 8; }
  }
  if (mybatch >= 0) {
    const float disc = sqrtf(fmaxf(scale * myval, 1e-12f));
    // All discrepancies > 0, so float order == int order of the bit patterns.
    atomicMin(out_bits + (size_t)mybatch * kNSh + sh, __float_as_int(disc));
  }
}

extern "C" void kernel_launch(void* const* d_in, const int* in_sizes, int n_in,
                              void* d_out, int out_size, void* d_ws, size_t ws_size,
                              hipStream_t stream) {
  (void)in_sizes; (void)n_in; (void)d_ws; (void)ws_size;
  const float* times     = (const float*)d_in[0];
  const float* path      = (const float*)d_in[1];
  const float* lengths   = (const float*)d_in[2];
  const float* shapelets = (const float*)d_in[3];
  int* out = (int*)d_out;

  shapelet_init_kernel<<<(out_size + 255) / 256, 256, 0, stream>>>(out, out_size);

  dim3 grid(2, kS, kNSh);   // 8192 independent waves
  shapelet_wmma_kernel<<<grid, 32, 0, stream>>>(times, path, lengths, shapelets, out);
}